// ProbAttention_58909771432892
// MI455X (gfx1250) — compile-verified
//
#include <hip/hip_runtime.h>
#include <hip/hip_bf16.h>

typedef float v2f __attribute__((ext_vector_type(2)));
typedef float v8f __attribute__((ext_vector_type(8)));

__device__ __forceinline__ v8f wmma_f32_4(v2f a, v2f b, v8f c) {
    // D = A(16x4,f32) * B(4x16,f32) + C(16x16,f32)
    return __builtin_amdgcn_wmma_f32_16x16x4_f32(
        false, a, false, b, (short)0, c, false, false);
}

#define NEG_BIG (-3.4e38f)
#define SSTR 1028   // score-row stride (words): bank = 4*row+col -> conflict-free frags
#define KSTR 68     // K/V staging row stride (words): conflict-free B-fragments

// ---------------------------------------------------------------------------
// Kernel 1: M[b,h,l] = max_s(Q[l]·K[sidx[l,s]]) - sum_s(...)/L
// One wave32 per query; lane owns 2 of the 64 dims.
// ---------------------------------------------------------------------------
__global__ void pa_msample(const float* __restrict__ Q, const float* __restrict__ K,
                           const int* __restrict__ sidx, float* __restrict__ Mbuf,
                           int B, int L, int H, int S) {
    int w    = (int)((blockIdx.x * blockDim.x + threadIdx.x) >> 5);
    int lane = threadIdx.x & 31;
    if (w >= B * H * L) return;
    int l  = w % L;
    int bh = w / L;
    int h  = bh % H;
    int b  = bh / H;

    const float* qp = Q + (((size_t)b * L + l) * H + h) * 64 + 2 * lane;
    float qx = qp[0], qy = qp[1];
    float mx = NEG_BIG, sm = 0.f;
    for (int s = 0; s < S; ++s) {
        int idx = sidx[l * S + s];
        const float* kp = K + (((size_t)b * L + idx) * H + h) * 64 + 2 * lane;
        float p = qx * kp[0] + qy * kp[1];
        for (int m = 1; m < 32; m <<= 1) p += __shfl_xor(p, m, 32);
        mx = fmaxf(mx, p);
        sm += p;
    }
    if (lane == 0) Mbuf[w] = mx - sm / (float)L;
}

// ---------------------------------------------------------------------------
// Kernel 2: per-(b,h) top-S of M over L, lower-index tie-break (jax top_k)
// ---------------------------------------------------------------------------
__global__ void pa_topk(const float* __restrict__ Mbuf, int* __restrict__ Mtop,
                        int L, int S) {
    int bh = blockIdx.x;
    int t  = threadIdx.x;  // 256 threads
    __shared__ float vs[1024];
    __shared__ float rv[256];
    __shared__ int   ri[256];
    for (int i = t; i < L; i += 256) vs[i] = Mbuf[(size_t)bh * L + i];
    __syncthreads();
    for (int k = 0; k < S; ++k) {
        float bv = NEG_BIG;
        int   bi = 0x7fffffff;
        for (int i = t; i < L; i += 256) {
            float x = vs[i];
            if (x > bv || (x == bv && i < bi)) { bv = x; bi = i; }
        }
        rv[t] = bv; ri[t] = bi;
        __syncthreads();
        for (int ofs = 128; ofs > 0; ofs >>= 1) {
            if (t < ofs) {
                float ov = rv[t + ofs]; int oi = ri[t + ofs];
                if (ov > rv[t] || (ov == rv[t] && oi < ri[t])) { rv[t] = ov; ri[t] = oi; }
            }
            __syncthreads();
        }
        if (t == 0) { Mtop[bh * S + k] = ri[0]; vs[ri[0]] = NEG_BIG; }
        __syncthreads();
    }
}

// ---------------------------------------------------------------------------
// Kernel 3: out[b,l,h,d] = cumsum_l(V[b,:,h,d]) / (l+1)  (baseline context)
// 1024 threads = 16 chunks x 64 dims, chunked scan.
// ---------------------------------------------------------------------------
__global__ void pa_cumavg(const float* __restrict__ V, float* __restrict__ out,
                          int B, int L, int H) {
    int bh = blockIdx.x;
    int b = bh / H, h = bh % H;
    int t = threadIdx.x;         // 1024
    int d = t & 63;
    int chunk = t >> 6;          // 0..15
    int CH = L >> 4;             // 64
    __shared__ float cs[16 * 64];
    size_t base   = (((size_t)b * L) * H + h) * 64 + d;
    size_t stride = (size_t)H * 64;
    int l0 = chunk * CH;
    float s = 0.f;
    for (int i = 0; i < CH; ++i) s += V[base + (size_t)(l0 + i) * stride];
    cs[chunk * 64 + d] = s;
    __syncthreads();
    float acc = 0.f;
    for (int c = 0; c < chunk; ++c) acc += cs[c * 64 + d];
    for (int i = 0; i < CH; ++i) {
        int l = l0 + i;
        acc += V[base + (size_t)l * stride];
        out[base + (size_t)l * stride] = acc / (float)(l + 1);
    }
}

// ---------------------------------------------------------------------------
// Kernel 4: attention over the S selected queries with the whole score matrix
// resident in the 320KB WGP LDS (CDNA5). fp32 WMMA 16x16x4 throughout.
// grid = B*H, block = NW*32 (NW = ceil(S/16) waves, each owns 16 query rows).
// Dynamic LDS layout (floats):
//   Slds[ROWS * SSTR]   scores -> exp(scores - rowmax)
//   KVb [64 * KSTR]     staging for K (phase A) then V (phase C)
//   Lsum[ROWS]          row sums of exp
//   Rm  [blockDim]      reduce scratch (max)
//   Rs  [blockDim]      reduce scratch (sum)
// ---------------------------------------------------------------------------
__global__ void pa_topattn(const float* __restrict__ Q, const float* __restrict__ K,
                           const float* __restrict__ V, const int* __restrict__ Mtop,
                           float* __restrict__ out, int B, int L, int H, int S) {
    extern __shared__ __align__(16) float smem[];
    const int bh   = blockIdx.x;
    const int b    = bh / H, h = bh % H;
    const int tid  = threadIdx.x;
    const int wave = tid >> 5;
    const int lane = tid & 31;
    const int hi   = lane >> 4;   // lane-half: K offset +2 / C-row M+8
    const int lm   = lane & 15;   // M for A-frags, N for B-frags
    const int ROWS = (blockDim.x >> 5) * 16;

    float* Slds = smem;
    float* KVb  = Slds + (size_t)ROWS * SSTR;
    float* Lsum = KVb + 64 * KSTR;
    float* Rm   = Lsum + ROWS;
    float* Rs   = Rm + blockDim.x;

    const float scale = 0.125f;   // 1/sqrt(64)
    const size_t kvbase = (((size_t)b * L) * H + h) * 64;
    const size_t rowstr = (size_t)H * 64;

    // Load this wave's 16 Q_reduce rows as A fragments (16 K-slices of 4).
    int u  = wave * 16 + lm;
    int uc = (u < S) ? u : 0;
    int qrow = Mtop[bh * S + uc];
    const float* qptr = Q + (((size_t)b * L + qrow) * H + h) * 64;
    float qa[16][2];
#pragma unroll
    for (int t = 0; t < 16; ++t) {
        qa[t][0] = qptr[4 * t + 2 * hi];
        qa[t][1] = qptr[4 * t + 1 + 2 * hi];
    }
    v8f zero8 = {0.f, 0.f, 0.f, 0.f, 0.f, 0.f, 0.f, 0.f};

    // ---------------- Phase A: scores -> Slds --------------------------------
    for (int k0 = 0; k0 < L; k0 += 64) {
        __syncthreads();
        for (int idx = tid; idx < 64 * 16; idx += blockDim.x) {
            int row = idx >> 4, c4 = idx & 15;
            const float* src = K + kvbase + (size_t)(k0 + row) * rowstr;
            *((float4*)(KVb + row * KSTR) + c4) = *((const float4*)src + c4);
            if (k0 + 64 < L)
                __builtin_prefetch(K + kvbase + (size_t)(k0 + 64 + row) * rowstr, 0, 1);
        }
        __syncthreads();
#pragma unroll
        for (int nn = 0; nn < 4; ++nn) {          // 16-key subtiles
            v8f c = zero8;
#pragma unroll
            for (int t = 0; t < 16; ++t) {
                v2f a = {qa[t][0], qa[t][1]};
                const float* kr = KVb + (nn * 16 + lm) * KSTR;
                v2f bb = {kr[4 * t + 2 * hi], kr[4 * t + 1 + 2 * hi]};
                c = wmma_f32_4(a, bb, c);
            }
            int col = k0 + nn * 16 + lm;
#pragma unroll
            for (int r = 0; r < 8; ++r)           // C elem (M=r+8*hi, N=lm)
                Slds[(size_t)(wave * 16 + r + 8 * hi) * SSTR + col] = c[r] * scale;
        }
    }
    __syncthreads();

    // ---------------- Phase B: softmax (two passes, no shuffles) ---------------
    {
        int row  = tid >> 1;              // 2 threads per row
        int half = tid & 1;
        float* sp = Slds + (size_t)row * SSTR + half * (L / 2);
        float m = NEG_BIG;
        for (int i = 0; i < L / 2; ++i) m = fmaxf(m, sp[i]);
        Rm[tid] = m;
        __syncthreads();
        float mrow = fmaxf(Rm[row * 2], Rm[row * 2 + 1]);
        float ssum = 0.f;
        for (int i = 0; i < L / 2; ++i) {
            float e = __expf(sp[i] - mrow);
            sp[i] = e;
            ssum += e;
        }
        Rs[tid] = ssum;
        __syncthreads();
        if (half == 0) Lsum[row] = Rs[row * 2] + Rs[row * 2 + 1];
    }

    // ---------------- Phase C: O = P * V -------------------------------------
    v8f O[4];
#pragma unroll
    for (int j = 0; j < 4; ++j) O[j] = zero8;
    const float* prow = Slds + (size_t)(wave * 16 + lm) * SSTR;

    for (int k0 = 0; k0 < L; k0 += 64) {
        __syncthreads();
        for (int idx = tid; idx < 64 * 16; idx += blockDim.x) {
            int row = idx >> 4, c4 = idx & 15;
            const float* src = V + kvbase + (size_t)(k0 + row) * rowstr;
            *((float4*)(KVb + row * KSTR) + c4) = *((const float4*)src + c4);
            if (k0 + 64 < L)
                __builtin_prefetch(V + kvbase + (size_t)(k0 + 64 + row) * rowstr, 0, 1);
        }
        __syncthreads();
#pragma unroll
        for (int kk = 0; kk < 64; kk += 4) {      // K-slices of 4 keys
            v2f a = {prow[k0 + kk + 2 * hi], prow[k0 + kk + 1 + 2 * hi]};
#pragma unroll
            for (int j = 0; j < 4; ++j) {
                v2f bb = {KVb[(kk + 2 * hi) * KSTR + j * 16 + lm],
                          KVb[(kk + 1 + 2 * hi) * KSTR + j * 16 + lm]};
                O[j] = wmma_f32_4(a, bb, O[j]);
            }
        }
    }

    // ---------------- Epilogue: normalize + scatter --------------------------
#pragma unroll
    for (int r = 0; r < 8; ++r) {
        int row = r + 8 * hi;
        int uu  = wave * 16 + row;
        if (uu >= S) continue;
        int l = Mtop[bh * S + uu];
        float inv = 1.0f / Lsum[wave * 16 + row];
        float* op = out + (((size_t)b * L + l) * H + h) * 64;
#pragma unroll
        for (int j = 0; j < 4; ++j) op[j * 16 + lm] = O[j][r] * inv;
    }
}

// ---------------------------------------------------------------------------
extern "C" void kernel_launch(void* const* d_in, const int* in_sizes, int n_in,
                              void* d_out, int out_size, void* d_ws, size_t ws_size,
                              hipStream_t stream) {
    const float* Q   = (const float*)d_in[0];
    const float* K   = (const float*)d_in[1];
    const float* V   = (const float*)d_in[2];
    const int* sidx  = (const int*)d_in[3];
    // d_in[4] = attn_mask scalar, unused.

    const int B = 8, L = 1024, H = 8;
    const int S = in_sizes[3] / L;   // 35

    float* Mbuf = (float*)d_ws;
    int*   Mtop = (int*)((char*)d_ws + sizeof(float) * (size_t)B * H * L);

    int nwaves = B * H * L;
    pa_msample<<<(nwaves * 32 + 255) / 256, 256, 0, stream>>>(Q, K, sidx, Mbuf, B, L, H, S);
    pa_topk<<<B * H, 256, 0, stream>>>(Mbuf, Mtop, L, S);
    pa_cumavg<<<B * H, 1024, 0, stream>>>(V, (float*)d_out, B, L, H);

    int NW   = (S + 15) / 16;        // 3 waves -> 48 rows
    int ROWS = NW * 16;
    size_t smem_floats = (size_t)ROWS * SSTR      // score matrix in LDS
                       + 64 * KSTR                // K/V staging chunk
                       + ROWS                     // row sums
                       + 2 * (size_t)(NW * 32);   // reduce scratch
    pa_topattn<<<B * H, NW * 32, smem_floats * sizeof(float), stream>>>(
        Q, K, V, Mtop, (float*)d_out, B, L, H, S);
}